// NCC_88192858456237
// MI455X (gfx1250) — compile-verified
//
#include <hip/hip_runtime.h>

// ---------------------------------------------------------------------------
// Local NCC loss, 9^3 window, fused separable box-sums via V_WMMA_F32_16X16X4_F32
//
// Roofline: 78.6 MB compulsory HBM reads -> ~3.4 us floor @ 23.3 TB/s.
// Working set < 192 MB L2, so the 3.375x halo redundancy (24^3 load per 16^3
// output tile) is L2-resident. Each 1D 9-tap box sum over a 24-long padded
// line producing 16 outputs is computed on the matrix unit as
//   out[16x16] = Band[16x24] x In[24x16]   (6 chained f32 16x16x4 WMMAs).
//
// All tile indices (wave id, channel, line base, h-slice) are forced into
// SGPRs via readfirstlane so every loop/branch is scalar: EXEC stays all-ones
// around the WMMAs (ISA requirement) and no per-lane address math is wasted.
// ---------------------------------------------------------------------------

typedef float v2f __attribute__((ext_vector_type(2)));
typedef float v8f __attribute__((ext_vector_type(8)));

namespace {
constexpr int WIN   = 9;
constexpr int HALO  = 4;        // WIN/2
constexpr int T     = 16;       // output tile edge
constexpr int THL   = 24;       // T + 2*HALO (padded tile edge)
constexpr int NCH   = 5;        // I, J, I*I, J*J, I*J

constexpr int DD = 160, HH = 192, WW = 160;
constexpr int BATCH = 2;
constexpr long long VOL = (long long)DD * HH * WW;
constexpr int TD = DD / T, THt = HH / T, TW = WW / T;   // 10, 12, 10
constexpr int NBLK = TW * THt * TD * BATCH;             // 2400

// LDS layout (floats). h5 overlays the sI/sJ region after stage 1.
constexpr int SI_OFF = 0;
constexpr int SJ_OFF = THL * THL * THL;                 // 13824
constexpr int H5_OFF = 0;
constexpr int H5_PER = THL * T * T;                     // 6144 per channel
constexpr int W5_OFF = NCH * H5_PER;                    // 30720 (>= 27648)
constexpr int W5_PER = THL * THL * T;                   // 9216 per channel
constexpr int LDS_FLOATS = W5_OFF + NCH * W5_PER;       // 76800 = 300 KB
}

// D = Band(16x24) x B(24x16) using 6 accumulating f32 16x16x4 WMMAs.
// A layout (32-bit 16x4): m=lane&15; VGPR0 holds K=kb, VGPR1 K=kb+1, kb=2*(lane>>4).
// B layout (32-bit 4x16): n=lane&15; same K split across lane halves.
template <typename LoadB>
__device__ __forceinline__ v8f band_apply(const v2f* Aband, int lane, LoadB loadB) {
  v8f acc = {};
  const int n  = lane & 15;
  const int kb = (lane >> 4) << 1;
#pragma unroll
  for (int c = 0; c < 6; ++c) {
    const int k = 4 * c + kb;
    v2f b;
    b.x = loadB(k,     n);
    b.y = loadB(k + 1, n);
    acc = __builtin_amdgcn_wmma_f32_16x16x4_f32(false, Aband[c], false, b,
                                                (short)0, acc, false, false);
  }
  return acc;
}

__global__ __launch_bounds__(512, 1)
void ncc_tile_kernel(const float* __restrict__ gI, const float* __restrict__ gJ,
                     float* __restrict__ partial) {
  __shared__ float lds[LDS_FLOATS];
  const int tid  = threadIdx.x;
  const int lane = tid & 31;
  // Wave index is uniform within a wave: pin it to an SGPR so all tile loops
  // and channel selects compile to scalar branches (EXEC untouched -> WMMA-safe).
  const int wave = __builtin_amdgcn_readfirstlane(tid >> 5);   // 0..15

  const int b  = blockIdx.z / TD;
  const int dz = blockIdx.z % TD;
  const int d0 = dz * T - HALO;
  const int h0 = blockIdx.y * T - HALO;
  const int w0 = blockIdx.x * T - HALO;
  const float* Ib = gI + (long long)b * VOL;
  const float* Jb = gJ + (long long)b * VOL;

  // ---- load 24^3 zero-padded halo tiles of I and J (27 iters, exact) ----
  for (int idx = tid; idx < THL * THL * THL; idx += 512) {
    const int lw = idx % THL;
    const int lh = (idx / THL) % THL;
    const int ld = idx / (THL * THL);
    const int gd = d0 + ld, gh = h0 + lh, gw = w0 + lw;
    float vi = 0.f, vj = 0.f;
    if ((unsigned)gd < (unsigned)DD && (unsigned)gh < (unsigned)HH &&
        (unsigned)gw < (unsigned)WW) {
      const long long g = ((long long)gd * HH + gh) * (long long)WW + gw;
      vi = Ib[g];
      vj = Jb[g];
    }
    lds[SI_OFF + idx] = vi;
    lds[SJ_OFF + idx] = vj;
  }

  // ---- constant band matrix in WMMA A layout: A[m][k] = (m <= k <= m+8) ----
  v2f Aband[6];
  {
    const int m  = lane & 15;
    const int kb = (lane >> 4) << 1;
#pragma unroll
    for (int c = 0; c < 6; ++c) {
      const int k0 = 4 * c + kb;
      Aband[c].x = (k0     >= m && k0     <= m + 8) ? 1.f : 0.f;
      Aband[c].y = (k0 + 1 >= m && k0 + 1 <= m + 8) ? 1.f : 0.f;
    }
  }
  __syncthreads();

  // ---- stage 1: box sum along W, 5 channels (WMMA). 576 (d,h) lines/ch ----
  for (int u = wave; u < NCH * 36; u += 16) {      // scalar loop
    const int ch = u / 36;                         // SGPR
    const int Lb = (u % 36) * 16;                  // SGPR: first of 16 lines
    v8f acc;
    switch (ch) {                                  // scalar branch, specialized
      case 0:
        acc = band_apply(Aband, lane, [&](int k, int n) {
          return lds[SI_OFF + (Lb + n) * THL + k];
        });
        break;
      case 1:
        acc = band_apply(Aband, lane, [&](int k, int n) {
          return lds[SJ_OFF + (Lb + n) * THL + k];
        });
        break;
      case 2:
        acc = band_apply(Aband, lane, [&](int k, int n) {
          const float a = lds[SI_OFF + (Lb + n) * THL + k];
          return a * a;
        });
        break;
      case 3:
        acc = band_apply(Aband, lane, [&](int k, int n) {
          const float a = lds[SJ_OFF + (Lb + n) * THL + k];
          return a * a;
        });
        break;
      default:
        acc = band_apply(Aband, lane, [&](int k, int n) {
          const int i = (Lb + n) * THL + k;
          return lds[SI_OFF + i] * lds[SJ_OFF + i];
        });
        break;
    }
    // D layout: VGPR r, lane<16 -> (w=r, line=Lb+lane); lane>=16 -> (w=r+8)
    const int n   = lane & 15;
    const int mof = (lane >> 4) * 8;
    float* dst = &lds[W5_OFF + (ch * (THL * THL) + Lb + n) * T + mof];
#pragma unroll
    for (int r = 0; r < 8; ++r) dst[r] = acc[r];
  }
  __syncthreads();   // w5 complete; sI/sJ dead -> h5 may overlay them

  // ---- stage 2: box sum along H (WMMA). w5[ch][d][h24][w16] -> h5[ch][d][h16][w16]
  for (int u = wave; u < NCH * THL; u += 16) {     // scalar loop, 120 tiles
    const int ch = u / THL;                        // SGPR
    const int d  = u % THL;                        // SGPR
    v8f acc = band_apply(Aband, lane, [&](int k, int n) {
      return lds[W5_OFF + (ch * (THL * THL) + d * THL + k) * T + n];
    });
    const int n   = lane & 15;
    const int mof = (lane >> 4) * 8;
#pragma unroll
    for (int r = 0; r < 8; ++r)
      lds[H5_OFF + ch * H5_PER + (d * T + mof + r) * T + n] = acc[r];
  }
  __syncthreads();

  // ---- stage 3: box sum along D (WMMA) + cc. wave w owns h-slice w ----
  const int h = wave;                              // SGPR
  v8f S[NCH];
#pragma unroll
  for (int ch = 0; ch < NCH; ++ch) {
    S[ch] = band_apply(Aband, lane, [&](int k, int n) {
      return lds[H5_OFF + ch * H5_PER + (k * T + h) * T + n];
    });
  }
  const float winf = (float)(WIN * WIN * WIN);
  const float inv  = 1.f / winf;
  float local = 0.f;
#pragma unroll
  for (int r = 0; r < 8; ++r) {
    const float Is = S[0][r], Js = S[1][r];
    const float I2 = S[2][r], J2 = S[3][r], IJ = S[4][r];
    const float uI = Is * inv, uJ = Js * inv;
    const float cross = IJ - uJ * Is - uI * Js + uI * uJ * winf;
    const float Iv = I2 - 2.f * uI * Is + uI * uI * winf;
    const float Jv = J2 - 2.f * uJ * Js + uJ * uJ * winf;
    local += cross * cross / (Iv * Jv + 1e-5f);
  }

  // ---- block reduction (deterministic) ----
#pragma unroll
  for (int off = 16; off > 0; off >>= 1) local += __shfl_down(local, off, 32);
  __syncthreads();                        // all h5 reads done before LDS reuse
  if (lane == 0) lds[wave] = local;
  __syncthreads();
  if (tid == 0) {
    float s = 0.f;
#pragma unroll
    for (int i = 0; i < 16; ++i) s += lds[i];
    const int bid = (blockIdx.z * gridDim.y + blockIdx.y) * gridDim.x + blockIdx.x;
    partial[bid] = s;
  }
}

// Deterministic final reduction: -mean(cc). Avoids float atomics so graph
// replays are bitwise stable; fully overwrites d_out every call.
__global__ __launch_bounds__(256)
void ncc_finalize_kernel(const float* __restrict__ partial, float* __restrict__ out) {
  __shared__ float red[256];
  float s = 0.f;
  for (int i = threadIdx.x; i < NBLK; i += 256) s += partial[i];
  red[threadIdx.x] = s;
  __syncthreads();
  for (int k = 128; k > 0; k >>= 1) {
    if (threadIdx.x < k) red[threadIdx.x] += red[threadIdx.x + k];
    __syncthreads();
  }
  if (threadIdx.x == 0)
    out[0] = -red[0] / (float)((long long)BATCH * VOL);
}

extern "C" void kernel_launch(void* const* d_in, const int* in_sizes, int n_in,
                              void* d_out, int out_size, void* d_ws, size_t ws_size,
                              hipStream_t stream) {
  (void)in_sizes; (void)n_in; (void)out_size; (void)ws_size;
  const float* I = (const float*)d_in[0];   // y_true
  const float* J = (const float*)d_in[1];   // y_pred
  float* partial = (float*)d_ws;            // NBLK floats (9.6 KB)

  dim3 grid(TW, THt, TD * BATCH);           // (10, 12, 20) = 2400 blocks
  ncc_tile_kernel<<<grid, 512, 0, stream>>>(I, J, partial);
  ncc_finalize_kernel<<<1, 256, 0, stream>>>(partial, (float*)d_out);
}